// GraphSAGELayer_28613072126846
// MI455X (gfx1250) — compile-verified
//
#include <hip/hip_runtime.h>

#define N_NODES 100000
#define N_EDGES 1200000
#define IN_DIM  64
#define OUT_DIM 64
#define K_DIM   128           // concat width = 2*IN_DIM
#define LN_EPS  1e-5f

#define XPAD 130              // LDS row stride for the 32x128 concat tile (8B aligned rows)
#define WPAD 130              // LDS row stride for W[n][k]
#define HPAD 66               // LDS row stride for the 32x64 h tile

typedef __attribute__((ext_vector_type(2))) float v2f;
typedef __attribute__((ext_vector_type(8))) float v8f;

// ---------------- kernel 1: zero agg + deg workspace ----------------
__global__ void sage_zero_ws(float* __restrict__ ws, int n) {
  int i = blockIdx.x * blockDim.x + threadIdx.x;
  if (i < n) ws[i] = 0.0f;
}

// ---------------- kernel 2: scatter-add of neighbor features + degree ----------------
// block = 256 threads = 4 edges x 64 dims; per-edge reads/atomics are 256B contiguous.
__global__ void sage_scatter(const float* __restrict__ x,
                             const long long* __restrict__ ei,   // int64 [2, N_EDGES]
                             float* __restrict__ agg,
                             float* __restrict__ deg) {
  int t = threadIdx.x;
  long long e = (long long)blockIdx.x * 4 + (t >> 6);
  int d = t & 63;
  if (e < N_EDGES) {
    int dst = (int)ei[e];              // row 0: aggregation target i
    int src = (int)ei[N_EDGES + e];    // row 1: neighbor j
    atomicAdd(&agg[(long long)dst * IN_DIM + d], x[(long long)src * IN_DIM + d]);
    if (d == 0) atomicAdd(&deg[dst], 1.0f);
  }
}

// ---------------- kernel 3: fused [x|agg/deg] @ W^T + b, ReLU, LayerNorm ----------------
// 256 threads = 8 wave32, 32 nodes per block, 2x4 grid of 16x16 WMMA tiles, K=128 in steps of 4.
__global__ __launch_bounds__(256) void sage_gemm_ln(
    const float* __restrict__ x, const float* __restrict__ W,
    const float* __restrict__ b, const float* __restrict__ gamma,
    const float* __restrict__ beta, const float* __restrict__ agg,
    const float* __restrict__ deg, float* __restrict__ out) {
  __shared__ float sW[OUT_DIM * WPAD];   // W[n][k], padded
  __shared__ float sX[32 * XPAD];        // concat tile [m][k], padded
  __shared__ float sH[32 * HPAD];        // post-relu h tile
  __shared__ float sGB[2 * OUT_DIM];     // gamma | beta
  __shared__ float sScale[32];           // 1 / max(deg, 1)

  const int tid = threadIdx.x;
  const int node0 = blockIdx.x * 32;

  if (tid < 32) {
    int g = node0 + tid;
    float dg = (g < N_NODES) ? deg[g] : 1.0f;
    sScale[tid] = 1.0f / fmaxf(dg, 1.0f);
  }
  if (tid < 2 * OUT_DIM)
    sGB[tid] = (tid < OUT_DIM) ? gamma[tid] : beta[tid - OUT_DIM];

  // W global is [64][128] row-major; keep [n][k] layout (consecutive k -> b64 LDS reads).
  for (int i = tid; i < OUT_DIM * K_DIM; i += 256) {
    int n = i >> 7, k = i & 127;
    sW[n * WPAD + k] = W[i];
  }
  // x half of the concat tile
  for (int i = tid; i < 32 * IN_DIM; i += 256) {
    int m = i >> 6, c = i & 63;
    int g = node0 + m;
    sX[m * XPAD + c] = (g < N_NODES) ? x[(long long)g * IN_DIM + c] : 0.0f;
  }
  __syncthreads();  // sScale ready
  // agg half, scaled by 1/deg (mean aggregation)
  for (int i = tid; i < 32 * IN_DIM; i += 256) {
    int m = i >> 6, c = i & 63;
    int g = node0 + m;
    sX[m * XPAD + IN_DIM + c] =
        (g < N_NODES) ? agg[(long long)g * IN_DIM + c] * sScale[m] : 0.0f;
  }
  __syncthreads();

  const int wave = tid >> 5, lane = tid & 31;
  const int wm = wave >> 2, wn = wave & 3;        // 2 x 4 tile grid
  const int l16 = lane & 15, lh = lane >> 4;
  // ISA A layout (16x4 f32): lane L -> row M=L%16, K = v + 2*(L/16); B mirrored.
  const float* ax = &sX[(wm * 16 + l16) * XPAD + lh * 2];
  const float* bw = &sW[(wn * 16 + l16) * WPAD + lh * 2];

  v8f c = {};
#pragma unroll
  for (int k0 = 0; k0 < K_DIM; k0 += 4) {
    v2f a  = *(const v2f*)(ax + k0);   // ds_load_b64
    v2f bb = *(const v2f*)(bw + k0);   // ds_load_b64
    // 8 args: (neg_a, A, neg_b, B, c_mod, C, reuse_a, reuse_b)
    c = __builtin_amdgcn_wmma_f32_16x16x4_f32(false, a, false, bb, (short)0, c,
                                              false, false);
  }

  // bias + relu -> LDS. C layout: lane L, VGPR v -> (M = v + 8*(L/16), N = L%16).
  const int ncol = wn * 16 + l16;
  const float bias = b[ncol];
#pragma unroll
  for (int v = 0; v < 8; ++v) {
    int m = wm * 16 + lh * 8 + v;
    sH[m * HPAD + ncol] = fmaxf(c[v] + bias, 0.0f);
  }
  __syncthreads();

  // LayerNorm: 8 lanes per node, wave32 shfl_xor tree (masks < 8 stay in-group).
  int m = tid >> 3, r = tid & 7;
  float vals[8];
  float s = 0.0f, s2 = 0.0f;
#pragma unroll
  for (int q = 0; q < 8; ++q) {
    float h = sH[m * HPAD + r * 8 + q];
    vals[q] = h;
    s += h;
    s2 += h * h;
  }
#pragma unroll
  for (int off = 1; off < 8; off <<= 1) {
    s  += __shfl_xor(s,  off, 32);
    s2 += __shfl_xor(s2, off, 32);
  }
  float mu   = s * (1.0f / OUT_DIM);
  float var  = s2 * (1.0f / OUT_DIM) - mu * mu;
  float rstd = rsqrtf(var + LN_EPS);

  int g = node0 + m;
  if (g < N_NODES) {
    float o[8];
#pragma unroll
    for (int q = 0; q < 8; ++q) {
      int col = r * 8 + q;
      o[q] = (vals[q] - mu) * rstd * sGB[col] + sGB[OUT_DIM + col];
    }
    float4* dst = (float4*)&out[(long long)g * OUT_DIM + r * 8];  // 32B aligned
    dst[0] = make_float4(o[0], o[1], o[2], o[3]);
    dst[1] = make_float4(o[4], o[5], o[6], o[7]);
  }
}

extern "C" void kernel_launch(void* const* d_in, const int* in_sizes, int n_in,
                              void* d_out, int out_size, void* d_ws, size_t ws_size,
                              hipStream_t stream) {
  (void)in_sizes; (void)n_in; (void)out_size; (void)ws_size;
  const float*     x     = (const float*)d_in[0];
  const float*     W     = (const float*)d_in[1];
  const float*     b     = (const float*)d_in[2];
  const float*     gamma = (const float*)d_in[3];
  const float*     beta  = (const float*)d_in[4];
  const long long* ei    = (const long long*)d_in[5];   // int64 edge_index
  float* out = (float*)d_out;

  float* agg = (float*)d_ws;                         // [N_NODES * 64]
  float* deg = agg + (size_t)N_NODES * IN_DIM;       // [N_NODES]

  int ztotal = N_NODES * IN_DIM + N_NODES;
  sage_zero_ws<<<(ztotal + 255) / 256, 256, 0, stream>>>(agg, ztotal);
  sage_scatter<<<(N_EDGES + 3) / 4, 256, 0, stream>>>(x, ei, agg, deg);
  sage_gemm_ln<<<(N_NODES + 31) / 32, 256, 0, stream>>>(x, W, b, gamma, beta,
                                                        agg, deg, out);
}